// Attention_88871463289454
// MI455X (gfx1250) — compile-verified
//
#include <hip/hip_runtime.h>
#include <hip/hip_bf16.h>

// Paged-attention GQA decode for MI455X (gfx1250), fp32 via V_WMMA_F32_16X16X4_F32.
// Memory-bound (AI ~2 FLOP/B): stream K/V at full fp32 precision, flash-decode split.

#define NBATCH 16
#define NH     32
#define NKV    8
#define NG     4      // H / KVH
#define ND     128
#define PAGE   256
#define NBLK   16
#define ATT_SCALE 0.08838834764831845f
#define NPART  32     // partials per (b,kv) = 4 split-WGs * 8 waves

#define OPART_ELEMS (128 * NPART * NG * ND)   // 2,097,152 floats
#define ML_ELEMS    (128 * NPART * NG)        // 16,384 floats

typedef __attribute__((ext_vector_type(2))) float v2f;
typedef __attribute__((ext_vector_type(8))) float v8f;

// One row of online softmax over a 16-token tile.
// Scores for this row live on lanes 0-15 (token = lane&15); lanes 16-31 carry
// harmless garbage (zero-padded rows) that never feeds the reductions we use.
__device__ __forceinline__ void online_row(float sv, bool valid,
                                           float& m, float& l,
                                           float& alpha, float& p) {
  sv = valid ? sv : -3.0e38f;
  float x = sv;                      // max over the 16-lane group
  x = fmaxf(x, __shfl_xor(x, 1));
  x = fmaxf(x, __shfl_xor(x, 2));
  x = fmaxf(x, __shfl_xor(x, 4));
  x = fmaxf(x, __shfl_xor(x, 8));
  float tmax = __shfl(x, 0);         // broadcast row max (from lanes 0-15 group)
  float mnew = fmaxf(m, tmax);
  alpha = __expf(m - mnew);
  p = __expf(sv - mnew);             // masked lanes -> exp(-huge) == 0
  float srow = p;                    // sum over the 16-lane group
  srow += __shfl_xor(srow, 1);
  srow += __shfl_xor(srow, 2);
  srow += __shfl_xor(srow, 4);
  srow += __shfl_xor(srow, 8);
  float rsum = __shfl(srow, 0);
  l = l * alpha + rsum;
  m = mnew;
}

__global__ __launch_bounds__(256) void pa_split_kernel(
    const float* __restrict__ q,      // [B, H, D]
    const float* __restrict__ knew,   // [B, KVH, D]
    const float* __restrict__ vnew,   // [B, KVH, D]
    const float* __restrict__ kcache, // [NUM_SLOTS, KVH, D]
    const float* __restrict__ vcache, // [NUM_SLOTS, KVH, D]
    const int*   __restrict__ block_tables, // [B, NBLK]
    const int*   __restrict__ context_lens, // [B]
    float* __restrict__ Opart,        // [B*KVH, NPART, G, D]
    float* __restrict__ Mpart,        // [B*KVH, NPART, G]
    float* __restrict__ Lpart) {      // [B*KVH, NPART, G]
  const int bk   = blockIdx.x;        // 0..127  (b*KVH + kv)
  const int b    = bk >> 3;
  const int kv   = bk & 7;
  const int w    = threadIdx.x >> 5;  // wave in workgroup, 0..7
  const int lane = threadIdx.x & 31;
  const int ml   = lane & 15;         // matrix row/col index within fragment
  const int hi   = lane >> 4;         // lane half (selects K pair in fragments)
  const int pi   = blockIdx.y * 8 + w; // partial index 0..31

  const int ctx    = context_lens[b];
  const int last   = ctx - 1;         // position of the freshly-written token
  const int ntiles = (ctx + 15) >> 4;
  const int* btb   = block_tables + b * NBLK;
  const float* knrow = knew + ((size_t)b * NKV + kv) * ND;
  const float* vnrow = vnew + ((size_t)b * NKV + kv) * ND;

  // Stage scaled Q[4x128] in LDS, then build the 32 fp32 A-fragments (K=4 each).
  __shared__ float qs[NG * ND];
  for (int i = threadIdx.x; i < NG * ND; i += 256) {
    int r = i >> 7, d = i & 127;
    qs[i] = q[((size_t)b * NH + kv * NG + r) * ND + d] * ATT_SCALE;
  }
  __syncthreads();

  v2f qa[32];
#pragma unroll
  for (int kb = 0; kb < 32; ++kb) {
    int kd = kb * 4 + hi * 2;
    v2f a;
    a[0] = (ml < NG) ? qs[ml * ND + kd + 0] : 0.0f;  // rows 4..15 zero-padded
    a[1] = (ml < NG) ? qs[ml * ND + kd + 1] : 0.0f;
    qa[kb] = a;
  }

  float m0 = -3.0e38f, m1 = -3.0e38f, m2 = -3.0e38f, m3 = -3.0e38f;
  float l0 = 0.f, l1 = 0.f, l2 = 0.f, l3 = 0.f;
  v8f acc[8];
#pragma unroll
  for (int nc = 0; nc < 8; ++nc) acc[nc] = (v8f){0,0,0,0,0,0,0,0};

  for (int t = pi; t < ntiles; t += NPART) {
    const int t0 = t << 4;                            // tile start token
    const int sbase = btb[t0 >> 8] * PAGE + (t0 & 255); // slot of first token

    // ---- K tile: lane supplies B-fragment column (token t0+ml) ----
    const int tokK = t0 + ml;
    const float* krow = (tokK == last)
        ? knrow
        : kcache + ((size_t)(sbase + ml) * NKV + kv) * ND;
    const v2f* kr2 = (const v2f*)krow;

    if (t + NPART < ntiles) {                         // prefetch next K row
      int nt0 = (t + NPART) << 4;
      int nsb = btb[nt0 >> 8] * PAGE + (nt0 & 255);
      __builtin_prefetch(kcache + ((size_t)(nsb + ml) * NKV + kv) * ND, 0, 0);
    }

    // ---- S = Q * K^T : accumulate D in K=4 chunks ----
    v8f s = (v8f){0,0,0,0,0,0,0,0};
#pragma unroll
    for (int kb = 0; kb < 32; ++kb) {
      v2f bf = kr2[kb * 2 + hi];  // K[token ml][4kb + 2hi + {0,1}]
      s = __builtin_amdgcn_wmma_f32_16x16x4_f32(false, qa[kb], false, bf,
                                                (short)0, s, false, false);
    }

    // ---- online softmax on the 4 valid rows (C VGPRs 0..3, lanes 0-15) ----
    const bool valid = tokK < ctx;
    float a0, a1, a2, a3, p0, p1, p2, p3;
    online_row(s[0], valid, m0, l0, a0, p0);
    online_row(s[1], valid, m1, l1, a1, p1);
    online_row(s[2], valid, m2, l2, a2, p2);
    online_row(s[3], valid, m3, l3, a3, p3);

    // ---- redistribute P (row r on lane t) into A-fragment layout ----
    v2f pfrag[4];
#pragma unroll
    for (int kc = 0; kc < 4; ++kc) {
      v2f af;
#pragma unroll
      for (int j = 0; j < 2; ++j) {
        int sl = kc * 4 + hi * 2 + j;   // source lane = token within tile
        float t0v = __shfl(p0, sl);
        float t1v = __shfl(p1, sl);
        float t2v = __shfl(p2, sl);
        float t3v = __shfl(p3, sl);
        af[j] = (ml == 0) ? t0v : (ml == 1) ? t1v
              : (ml == 2) ? t2v : (ml == 3) ? t3v : 0.0f;
      }
      pfrag[kc] = af;
    }

    // ---- V row pointers for this lane's B-fragment rows ----
    const float* vrp[4][2];
#pragma unroll
    for (int kc = 0; kc < 4; ++kc)
#pragma unroll
      for (int j = 0; j < 2; ++j) {
        int off = kc * 4 + hi * 2 + j;
        int tk = t0 + off;
        vrp[kc][j] = (tk == last)
            ? vnrow
            : vcache + ((size_t)(sbase + off) * NKV + kv) * ND;
      }

    // ---- O = alpha*O + P*V over 8 chunks of 16 head-dims ----
#pragma unroll
    for (int nc = 0; nc < 8; ++nc) {
      v8f c = acc[nc];
      c[0] *= a0; c[1] *= a1; c[2] *= a2; c[3] *= a3;  // rows 4..7 stay zero
#pragma unroll
      for (int kc = 0; kc < 4; ++kc) {
        v2f bv;
        bv[0] = vrp[kc][0][nc * 16 + ml];
        bv[1] = vrp[kc][1][nc * 16 + ml];
        c = __builtin_amdgcn_wmma_f32_16x16x4_f32(false, pfrag[kc], false, bv,
                                                  (short)0, c, false, false);
      }
      acc[nc] = c;
    }
  }

  // ---- write per-wave partials (unnormalized O, running m/l) ----
  const size_t pb = (size_t)bk * NPART + pi;
  if (lane < 16) {
#pragma unroll
    for (int r = 0; r < NG; ++r)
#pragma unroll
      for (int nc = 0; nc < 8; ++nc)
        Opart[(pb * NG + r) * ND + nc * 16 + ml] = acc[nc][r];
  }
  if (lane == 0) {
    Mpart[pb * NG + 0] = m0; Mpart[pb * NG + 1] = m1;
    Mpart[pb * NG + 2] = m2; Mpart[pb * NG + 3] = m3;
    Lpart[pb * NG + 0] = l0; Lpart[pb * NG + 1] = l1;
    Lpart[pb * NG + 2] = l2; Lpart[pb * NG + 3] = l3;
  }
}

__global__ __launch_bounds__(256) void pa_reduce_kernel(
    const float* __restrict__ Opart,
    const float* __restrict__ Mpart,
    const float* __restrict__ Lpart,
    float* __restrict__ out) {
  const int e = blockIdx.x * 256 + threadIdx.x;   // 0..65535
  const int d  = e & 127;
  const int r  = (e >> 7) & 3;
  const int bk = e >> 9;

  float M = -3.0e38f;
#pragma unroll
  for (int pi = 0; pi < NPART; ++pi)
    M = fmaxf(M, Mpart[((size_t)bk * NPART + pi) * NG + r]);

  float L = 0.f, O = 0.f;
#pragma unroll
  for (int pi = 0; pi < NPART; ++pi) {
    size_t idx = (size_t)bk * NPART + pi;
    float f = __expf(Mpart[idx * NG + r] - M);
    L += Lpart[idx * NG + r] * f;
    O += Opart[(idx * NG + r) * ND + d] * f;
  }

  const int b = bk >> 3, kv = bk & 7;
  out[((size_t)b * NH + kv * NG + r) * ND + d] = O / L;
}

extern "C" void kernel_launch(void* const* d_in, const int* in_sizes, int n_in,
                              void* d_out, int out_size, void* d_ws, size_t ws_size,
                              hipStream_t stream) {
  const float* q      = (const float*)d_in[0];
  const float* knew   = (const float*)d_in[1];
  const float* vnew   = (const float*)d_in[2];
  const float* kcache = (const float*)d_in[3];
  const float* vcache = (const float*)d_in[4];
  // d_in[5] = slot_mapping (unused: slot of the new token == position ctx-1,
  // and we redirect reads of token ctx-1 to knew/vnew instead of mutating caches)
  const int* block_tables = (const int*)d_in[6];
  const int* context_lens = (const int*)d_in[7];

  float* Opart = (float*)d_ws;
  float* Mpart = Opart + OPART_ELEMS;
  float* Lpart = Mpart + ML_ELEMS;

  dim3 gridA(NBATCH * NKV, 4);   // 128 (b,kv) pairs x 4 context splits
  pa_split_kernel<<<gridA, 256, 0, stream>>>(
      q, knew, vnew, kcache, vcache, block_tables, context_lens,
      Opart, Mpart, Lpart);

  pa_reduce_kernel<<<(NBATCH * NH * ND) / 256, 256, 0, stream>>>(
      Opart, Mpart, Lpart, (float*)d_out);
}